// HeteroGraphTransformer_11321533792645
// MI455X (gfx1250) — compile-verified
//
#include <hip/hip_runtime.h>
#include <cstddef>

// ---------------------------------------------------------------------------
// Problem constants (match reference)
// ---------------------------------------------------------------------------
#define GN   16384        // nodes per type
#define GHD  64           // hidden dim
#define GHC  256          // heads * per-head dim (4 * 64)
#define GE   262144       // edges per edge type
#define GL   2            // layers
#define MTIL 4            // vertical 16-row tiles per wave in the GEMM

typedef float v2f __attribute__((ext_vector_type(2)));
typedef float v8f __attribute__((ext_vector_type(8)));

// ---------------------------------------------------------------------------
// fp32 WMMA GEMM:  out[M x Nc] = A[M x K] @ W[K x Nc] + bias (+ extras)
//
// One wave32 owns one 16-wide column stripe (tn) and MTIL vertical 16x16
// tiles. B fragments for the full K extent are loaded ONCE into registers
// (K/4 x v2f), then reused across the MTIL tiles -> 4x fewer B loads and no
// per-iteration address math (K is a template constant, loops fully unroll).
//
// Fragment layouts per CDNA5 ISA 7.12.2 (V_WMMA_F32_16X16X4_F32):
//   A (16x4, 2 VGPR):  lanes 0-15 -> M=lane, {K=k0,k0+1}; lanes 16-31 -> {k0+2,k0+3}
//   B (4x16, 2 VGPR):  lanes 0-15 -> N=lane, {K=k0,k0+1}; lanes 16-31 -> {k0+2,k0+3}
//   C/D (16x16, 8 VGPR): vgpr i, lanes 0-15 -> M=i, N=lane; lanes 16-31 -> M=8+i
// ---------------------------------------------------------------------------
template <int K, bool GELU>
__global__ void gemm_wmma_f32(const float* __restrict__ A, int lda,
                              const float* __restrict__ W, int ldb,
                              const float* __restrict__ bias,
                              float* __restrict__ out, int ldc,
                              int tiles_n,
                              const float* __restrict__ extra_col,   // per-column add (may be null)
                              const float* __restrict__ extra_rc)    // per row*ldc+col add (may be null)
{
    constexpr int KS = K / 4;

    const int wave = (blockIdx.x * blockDim.x + threadIdx.x) >> 5;
    const int lane = threadIdx.x & 31;
    const int tg   = wave / tiles_n;          // group of MTIL m-tiles
    const int tn   = wave % tiles_n;

    const int half = lane >> 4;               // 0: K pair {0,1}, 1: K pair {2,3}
    const int l15  = lane & 15;
    const int col  = tn * 16 + l15;

    // ---- preload B fragments for the entire K extent (kept in VGPRs)
    v2f bfrag[KS];
    {
        const float* wcol = W + (size_t)(half * 2) * ldb + col;
#pragma unroll
        for (int ks = 0; ks < KS; ++ks) {
            bfrag[ks].x = wcol[(size_t)(4 * ks + 0) * ldb];
            bfrag[ks].y = wcol[(size_t)(4 * ks + 1) * ldb];
        }
    }

    const float bc = bias ? bias[col] : 0.f;
    const float ec = extra_col ? extra_col[col] : 0.f;
    const int row_base = tg * (16 * MTIL);

#pragma unroll
    for (int mt = 0; mt < MTIL; ++mt) {
        const int rowA = row_base + mt * 16 + l15;
        const float* arow = A + (size_t)rowA * lda + half * 2;

        // prefetch the next tile's A row a tile ahead (global_prefetch_b8 path)
        if (mt + 1 < MTIL)
            __builtin_prefetch(A + (size_t)(rowA + 16) * lda + half * 2, 0, 3);

        v8f acc = {0.f, 0.f, 0.f, 0.f, 0.f, 0.f, 0.f, 0.f};
#pragma unroll
        for (int ks = 0; ks < KS; ++ks) {
            v2f a;
            a.x = arow[4 * ks + 0];
            a.y = arow[4 * ks + 1];
            acc = __builtin_amdgcn_wmma_f32_16x16x4_f32(
                /*neg_a=*/false, a, /*neg_b=*/false, bfrag[ks],
                /*c_mod=*/(short)0, acc, /*reuse_a=*/false, /*reuse_b=*/false);
        }

        const int row0 = row_base + mt * 16 + half * 8;
#pragma unroll
        for (int i = 0; i < 8; ++i) {
            const int row = row0 + i;
            float val = acc[i] + bc + ec;
            if (extra_rc) val += extra_rc[(size_t)row * ldc + col];
            if (GELU) val = 0.5f * val * (1.f + erff(val * 0.70710678118654752f));
            out[(size_t)row * ldc + col] = val;
        }
    }
}

// ---------------------------------------------------------------------------
// Ordered-uint encoding for float atomic max
// ---------------------------------------------------------------------------
__device__ __forceinline__ unsigned ford(float f) {
    unsigned u = __float_as_uint(f);
    return (u & 0x80000000u) ? ~u : (u | 0x80000000u);
}
__device__ __forceinline__ float ford_inv(unsigned u) {
    u = (u & 0x80000000u) ? (u & 0x7FFFFFFFu) : ~u;
    return __uint_as_float(u);
}
#define FORD_NEG_INF 0x007FFFFFu   // ford(-inf)

// ---------------------------------------------------------------------------
// init: accum[N*256]=0, denom[N*4]=0, mmax[N*4]=ford(-inf)
// ---------------------------------------------------------------------------
__global__ void init_attn(float* __restrict__ accum, float* __restrict__ denom,
                          unsigned* __restrict__ mmax, int n)
{
    const int idx = blockIdx.x * blockDim.x + threadIdx.x;
    if (idx < n * 256) accum[idx] = 0.f;
    if (idx < n * 4) { denom[idx] = 0.f; mmax[idx] = FORD_NEG_INF; }
}

// ---------------------------------------------------------------------------
// Pass 1: per-edge logits (wave per edge; 8 lanes per head) + segment max
// 128-bit gathers: each lane reads 2 x float4 of q and k.
// ---------------------------------------------------------------------------
__global__ void edge_logits(const int* __restrict__ src, const int* __restrict__ dst,
                            const float* __restrict__ q, const float* __restrict__ k,
                            float* __restrict__ logit, unsigned* __restrict__ mmax, int ecnt)
{
    const int e    = (blockIdx.x * blockDim.x + threadIdx.x) >> 5;
    const int lane = threadIdx.x & 31;
    if (e >= ecnt) return;
    const int s = src[e], d = dst[e];
    const float4* qp = (const float4*)(q + (size_t)d * GHC + lane * 8);
    const float4* kp = (const float4*)(k + (size_t)s * GHC + lane * 8);
    const float4 q0 = qp[0], q1 = qp[1];
    const float4 k0 = kp[0], k1 = kp[1];
    float acc = q0.x * k0.x + q0.y * k0.y + q0.z * k0.z + q0.w * k0.w
              + q1.x * k1.x + q1.y * k1.y + q1.z * k1.z + q1.w * k1.w;
    acc += __shfl_xor(acc, 1, 8);
    acc += __shfl_xor(acc, 2, 8);
    acc += __shfl_xor(acc, 4, 8);
    if ((lane & 7) == 0) {
        const int h = lane >> 3;
        const float lg = acc * 0.125f;               // 1/sqrt(64)
        logit[(size_t)e * 4 + h] = lg;
        atomicMax(&mmax[(size_t)d * 4 + h], ford(lg));
    }
}

// ---------------------------------------------------------------------------
// Pass 2: exp(logit - max) + segment-sum denominator (thread per edge*head)
// logit and ex may alias (read-then-write same element).
// ---------------------------------------------------------------------------
__global__ void edge_exp(const int* __restrict__ dst,
                         const float* __restrict__ logit, float* __restrict__ ex,
                         const unsigned* __restrict__ mmax, float* __restrict__ denom,
                         int etot)
{
    const int idx = blockIdx.x * blockDim.x + threadIdx.x;
    if (idx >= etot) return;
    const int e = idx >> 2, h = idx & 3;
    const int d = dst[e];
    const float m = ford_inv(mmax[(size_t)d * 4 + h]);
    const float ev = expf(logit[idx] - m);
    ex[idx] = ev;
    atomicAdd(&denom[(size_t)d * 4 + h], ev);
}

// ---------------------------------------------------------------------------
// Pass 3: scatter alpha * v into accum (wave per edge, b128 gathers of v)
// ---------------------------------------------------------------------------
__global__ void edge_scatter(const int* __restrict__ src, const int* __restrict__ dst,
                             const float* __restrict__ ex, const float* __restrict__ denom,
                             const float* __restrict__ v, float* __restrict__ accum, int ecnt)
{
    const int e    = (blockIdx.x * blockDim.x + threadIdx.x) >> 5;
    const int lane = threadIdx.x & 31;
    if (e >= ecnt) return;
    const int s = src[e], d = dst[e];
    const int h = lane >> 3;
    const float alpha = ex[(size_t)e * 4 + h] / denom[(size_t)d * 4 + h];
    const float4* vp = (const float4*)(v + (size_t)s * GHC + lane * 8);
    const float4 v0 = vp[0], v1 = vp[1];
    float* op = accum + (size_t)d * GHC + lane * 8;
    atomicAdd(&op[0], alpha * v0.x);
    atomicAdd(&op[1], alpha * v0.y);
    atomicAdd(&op[2], alpha * v0.z);
    atomicAdd(&op[3], alpha * v0.w);
    atomicAdd(&op[4], alpha * v1.x);
    atomicAdd(&op[5], alpha * v1.y);
    atomicAdd(&op[6], alpha * v1.z);
    atomicAdd(&op[7], alpha * v1.w);
}

// ---------------------------------------------------------------------------
// Combine: head mean, beta gate (sigmoid of 192-d dot), accumulate into hn[d]
// One wave per dst node; each lane owns channels {lane, lane+32}.
// ---------------------------------------------------------------------------
__global__ void combine_gate(const float* __restrict__ accum, const float* __restrict__ r,
                             const float* __restrict__ Wb, float* __restrict__ hn, int ncnt)
{
    const int n    = (blockIdx.x * blockDim.x + threadIdx.x) >> 5;
    const int lane = threadIdx.x & 31;
    if (n >= ncnt) return;

    float outs[2], rs[2];
    float part = 0.f;
#pragma unroll
    for (int t = 0; t < 2; ++t) {
        const int c = lane + t * 32;
        const size_t base = (size_t)n * GHC;
        const float o = 0.25f * (accum[base + c] + accum[base + 64 + c] +
                                 accum[base + 128 + c] + accum[base + 192 + c]);
        const float rr = r[(size_t)n * GHD + c];
        outs[t] = o; rs[t] = rr;
        part += o * Wb[c] + rr * Wb[64 + c] + (o - rr) * Wb[128 + c];
    }
#pragma unroll
    for (int off = 1; off < 32; off <<= 1) part += __shfl_xor(part, off, 32);
    const float beta = 1.f / (1.f + expf(-part));
#pragma unroll
    for (int t = 0; t < 2; ++t) {
        const int c = lane + t * 32;
        hn[(size_t)n * GHD + c] += beta * rs[t] + (1.f - beta) * outs[t];
    }
}

// ---------------------------------------------------------------------------
// LayerNorm over 64 channels of (h + hn); one wave per node, 2 ch per lane.
// ---------------------------------------------------------------------------
__global__ void layernorm(const float* __restrict__ h, const float* __restrict__ hn,
                          const float* __restrict__ g, const float* __restrict__ b,
                          float* __restrict__ hr, int ncnt)
{
    const int n    = (blockIdx.x * blockDim.x + threadIdx.x) >> 5;
    const int lane = threadIdx.x & 31;
    if (n >= ncnt) return;
    const size_t base = (size_t)n * GHD;
    const float x0 = h[base + lane]      + hn[base + lane];
    const float x1 = h[base + lane + 32] + hn[base + lane + 32];
    float s  = x0 + x1;
    float s2 = x0 * x0 + x1 * x1;
#pragma unroll
    for (int off = 1; off < 32; off <<= 1) {
        s  += __shfl_xor(s,  off, 32);
        s2 += __shfl_xor(s2, off, 32);
    }
    const float mu   = s * (1.f / 64.f);
    const float var  = s2 * (1.f / 64.f) - mu * mu;
    const float rstd = rsqrtf(var + 1e-5f);
    hr[base + lane]      = (x0 - mu) * rstd * g[lane]      + b[lane];
    hr[base + lane + 32] = (x1 - mu) * rstd * g[lane + 32] + b[lane + 32];
}

// ---------------------------------------------------------------------------
// Host orchestration
// ---------------------------------------------------------------------------
extern "C" void kernel_launch(void* const* d_in, const int* in_sizes, int n_in,
                              void* d_out, int out_size, void* d_ws, size_t ws_size,
                              hipStream_t stream)
{
    (void)in_sizes; (void)n_in; (void)out_size; (void)ws_size;

    const float* x    = (const float*)d_in[0];
    const int*   ei   = (const int*)  d_in[1];
    const float* Win  = (const float*)d_in[2];
    const float* bin  = (const float*)d_in[3];
    const float* temb = (const float*)d_in[4];
    const float* pos  = (const float*)d_in[5];
    const float* Wq   = (const float*)d_in[6];
    const float* bq   = (const float*)d_in[7];
    const float* Wk   = (const float*)d_in[8];
    const float* bk   = (const float*)d_in[9];
    const float* Wv   = (const float*)d_in[10];
    const float* bv   = (const float*)d_in[11];
    const float* Wsk  = (const float*)d_in[12];
    const float* bsk  = (const float*)d_in[13];
    const float* Wb   = (const float*)d_in[14];
    const float* lng  = (const float*)d_in[15];
    const float* lnb  = (const float*)d_in[16];
    const float* fW1  = (const float*)d_in[17];
    const float* fb1  = (const float*)d_in[18];
    const float* fW2  = (const float*)d_in[19];
    const float* fb2  = (const float*)d_in[20];
    const float* Wout = (const float*)d_in[21];
    const float* bout = (const float*)d_in[22];
    float*       outp = (float*)d_out;

    const size_t N64  = (size_t)GN * GHD;    // 1,048,576
    const size_t N256 = (size_t)GN * GHC;    // 4,194,304
    const size_t E4   = (size_t)GE * 4;      // 1,048,576

    float* ws = (float*)d_ws;
    float* h0    = ws;                 float* h1  = ws + N64;
    float* hn0   = ws + 2 * N64;       float* hn1 = ws + 3 * N64;
    float* qb    = ws + 4 * N64;
    float* kb    = qb + N256;
    float* vb    = kb + N256;
    float* accum = vb + N256;
    float* exb   = accum + N256;                     // E*4 logits/ex
    unsigned* mmax = (unsigned*)(exb + E4);          // N*4
    float* denom = (float*)(mmax + (size_t)GN * 4);  // N*4
    float* rb    = denom + (size_t)GN * 4;           // N*64
    float* hrb   = rb + N64;                         // N*64
    float* midb  = qb;                               // alias: FFN hidden reuses q

    float* h[2]  = { h0, h1 };
    float* hn[2] = { hn0, hn1 };

    // GEMM launcher: M = GN always; one wave per (MTIL x 16)-row, 16-col stripe.
    auto gemm = [&](const float* A, int lda, const float* W, int ldb, const float* bias,
                    float* out, int ldc, int K, int Nc,
                    const float* ecol, const float* erc, bool gelu) {
        const int tiles_n = Nc / 16;
        const int waves   = (GN / (16 * MTIL)) * tiles_n;
        const int blocks  = (waves * 32) / 256;
        if (K == 64) {
            if (gelu)
                gemm_wmma_f32<64, true><<<blocks, 256, 0, stream>>>(A, lda, W, ldb, bias, out, ldc,
                                                                    tiles_n, ecol, erc);
            else
                gemm_wmma_f32<64, false><<<blocks, 256, 0, stream>>>(A, lda, W, ldb, bias, out, ldc,
                                                                     tiles_n, ecol, erc);
        } else { // K == 256 (FF2) - never needs GELU
            gemm_wmma_f32<256, false><<<blocks, 256, 0, stream>>>(A, lda, W, ldb, bias, out, ldc,
                                                                  tiles_n, ecol, erc);
        }
    };

    const int edge_waves_blocks = (GE * 32) / 256;   // wave-per-edge kernels
    const int node_waves_blocks = (GN * 32) / 256;   // wave-per-node kernels

    // ---- input projection: h[t] = x[t]@W_in[t] + b_in[t] + type_emb[t] + pos[t]
    for (int t = 0; t < 2; ++t)
        gemm(x + t * N64, GHD, Win + (size_t)t * GHD * GHD, GHD, bin + t * GHD,
             h[t], GHD, GHD, GHD, temb + t * GHD, pos + t * N64, false);

    // ---- layers
    for (int l = 0; l < GL; ++l) {
        hipMemsetAsync(hn[0], 0, N64 * sizeof(float), stream);
        hipMemsetAsync(hn[1], 0, N64 * sizeof(float), stream);

        for (int e = 0; e < 2; ++e) {
            const int s = (e == 0) ? 0 : 1;
            const int d = (e == 0) ? 1 : 0;
            const int* srcI = ei + (size_t)e * 2 * GE;
            const int* dstI = srcI + GE;
            const size_t w = (size_t)l * 2 + e;

            gemm(h[d], GHD, Wq + w * GHD * GHC, GHC, bq + w * GHC, qb, GHC, GHD, GHC, 0, 0, false);
            gemm(h[s], GHD, Wk + w * GHD * GHC, GHC, bk + w * GHC, kb, GHC, GHD, GHC, 0, 0, false);
            gemm(h[s], GHD, Wv + w * GHD * GHC, GHC, bv + w * GHC, vb, GHC, GHD, GHC, 0, 0, false);
            gemm(h[d], GHD, Wsk + w * GHD * GHD, GHD, bsk + w * GHD, rb, GHD, GHD, GHD, 0, 0, false);

            init_attn<<<(GN * 256) / 256, 256, 0, stream>>>(accum, denom, mmax, GN);
            edge_logits<<<edge_waves_blocks, 256, 0, stream>>>(srcI, dstI, qb, kb, exb, mmax, GE);
            edge_exp<<<(GE * 4) / 256, 256, 0, stream>>>(dstI, exb, exb, mmax, denom, GE * 4);
            edge_scatter<<<edge_waves_blocks, 256, 0, stream>>>(srcI, dstI, exb, denom, vb, accum, GE);
            combine_gate<<<node_waves_blocks, 256, 0, stream>>>(accum, rb, Wb + w * 192, hn[d], GN);
        }

        for (int t = 0; t < 2; ++t) {
            const size_t w = (size_t)l * 2 + t;
            layernorm<<<node_waves_blocks, 256, 0, stream>>>(h[t], hn[t],
                                                             lng + w * GHD, lnb + w * GHD, hrb, GN);
            gemm(hrb, GHD, fW1 + w * GHD * (4 * GHD), 4 * GHD, fb1 + w * (4 * GHD),
                 midb, 4 * GHD, GHD, 4 * GHD, 0, 0, true);                      // GELU fused
            gemm(midb, 4 * GHD, fW2 + w * (4 * GHD) * GHD, GHD, fb2 + w * GHD,
                 h[t], GHD, 4 * GHD, GHD, 0, hrb, false);                       // residual fused
        }
    }

    // ---- output projection
    for (int t = 0; t < 2; ++t)
        gemm(h[t], GHD, Wout + (size_t)t * GHD * GHD, GHD, bout + t * GHD,
             outp + t * N64, GHD, GHD, GHD, 0, 0, false);
}